// ChannelAttn_31671088840844
// MI455X (gfx1250) — compile-verified
//
#include <hip/hip_runtime.h>
#include <math.h>

// Channel attention (SE block) for x=[64,512,56,56] fp32.
// Bandwidth-bound on the global-average-pool (411 MB streamed once, ~17.6us
// at 23.3 TB/s). The tiny MLP (64x512x32 / 64x32x512) uses fp32 WMMA
// (v_wmma_f32_16x16x4_f32) to stay in full precision on the matrix pipe.

typedef __attribute__((ext_vector_type(2))) float v2f;
typedef __attribute__((ext_vector_type(4))) float v4f;
typedef __attribute__((ext_vector_type(8))) float v8f;

#define B_    64
#define C_    512
#define HW_   3136   // 56*56
#define CB_   32
#define BN_EPS 1e-5f

// ---------------------------------------------------------------------------
// Kernel 1: global average pool  x[B,C,H,W] -> g[B,C]
// One wave32 per (b,c) row; 128-bit non-temporal streaming loads.
// ---------------------------------------------------------------------------
__global__ void se_pool_kernel(const float* __restrict__ x,
                               float* __restrict__ g) {
    const int wave = threadIdx.x >> 5;
    const int lane = threadIdx.x & 31;
    const int row  = blockIdx.x * 8 + wave;          // row in [0, B*C)
    const v4f* __restrict__ xr = (const v4f*)x + (size_t)row * (HW_ / 4);

    float s = 0.0f;
    // 784 float4 per row; lanes stride by 32 -> 24/25 b128 loads per lane.
    for (int i = lane; i < HW_ / 4; i += 32) {
        v4f v = __builtin_nontemporal_load(xr + i);  // NT: x >> L2, stream it
        s += v.x + v.y + v.z + v.w;
    }
    // wave32 reduction
    #pragma unroll
    for (int off = 16; off > 0; off >>= 1)
        s += __shfl_xor(s, off, 32);
    if (lane == 0)
        g[row] = s * (1.0f / (float)HW_);
}

// ---------------------------------------------------------------------------
// Kernel 2a: h = ReLU(BN(g @ w1^T + b1))   [64,512]x[512,32] -> [64,32]
// One block, 8 waves; wave w owns tile (mt = w>>1, nt = w&1).
// A layout (16x4 f32): lanes 0-15 hold K=k,k+1 ; lanes 16-31 hold K=k+2,k+3.
// ---------------------------------------------------------------------------
__global__ void se_mlp1_kernel(const float* __restrict__ g,
                               const float* __restrict__ w1,
                               const float* __restrict__ b1,
                               const float* __restrict__ bn_gamma,
                               const float* __restrict__ bn_beta,
                               const float* __restrict__ bn_mean,
                               const float* __restrict__ bn_var,
                               float* __restrict__ h) {
    const int wave = threadIdx.x >> 5;
    const int lane = threadIdx.x & 31;
    const int mt = wave >> 1;            // 0..3  (M tile of 16)
    const int nt = wave & 1;             // 0..1  (N tile of 16)
    const int m0 = mt * 16 + (lane & 15);
    const int n0 = nt * 16 + (lane & 15);
    const int khalf = (lane >> 4) << 1;  // 0 for lanes 0-15, 2 for 16-31

    v8f acc = {};
    for (int k = 0; k < C_; k += 4) {
        v2f a, b;
        a.x = g[m0 * C_ + k + khalf];
        a.y = g[m0 * C_ + k + khalf + 1];
        // B[k][n] = w1[n][k]  (w1 is [32,512] row-major)
        b.x = w1[n0 * C_ + k + khalf];
        b.y = w1[n0 * C_ + k + khalf + 1];
        acc = __builtin_amdgcn_wmma_f32_16x16x4_f32(
            /*neg_a=*/false, a, /*neg_b=*/false, b,
            /*c_mod=*/(short)0, acc, /*reuse_a=*/false, /*reuse_b=*/false);
    }

    // Epilogue: this lane holds column n0, rows rbase..rbase+7.
    const float sc   = bn_gamma[n0] * rsqrtf(bn_var[n0] + BN_EPS);
    const float bias = b1[n0];
    const float mu   = bn_mean[n0];
    const float bt   = bn_beta[n0];
    const int rbase  = mt * 16 + ((lane >> 4) << 3);
    #pragma unroll
    for (int r = 0; r < 8; ++r) {
        float v = acc[r] + bias;
        v = (v - mu) * sc + bt;
        v = fmaxf(v, 0.0f);
        h[(rbase + r) * CB_ + n0] = v;
    }
}

// ---------------------------------------------------------------------------
// Kernel 2b: out = g * sigmoid(h @ w2^T + b2)   [64,32]x[32,512] -> [64,512]
// 128 tiles (4 M x 32 N); 16 blocks x 8 waves, one tile per wave.
// ---------------------------------------------------------------------------
__global__ void se_mlp2_kernel(const float* __restrict__ h,
                               const float* __restrict__ w2,
                               const float* __restrict__ b2,
                               const float* __restrict__ g,
                               float* __restrict__ out) {
    const int wave = threadIdx.x >> 5;
    const int lane = threadIdx.x & 31;
    const int tile = blockIdx.x * 8 + wave;   // 0..127
    const int mt = tile >> 5;                 // 0..3
    const int nt = tile & 31;                 // 0..31
    const int m0 = mt * 16 + (lane & 15);
    const int n0 = nt * 16 + (lane & 15);
    const int khalf = (lane >> 4) << 1;

    v8f acc = {};
    #pragma unroll
    for (int k = 0; k < CB_; k += 4) {
        v2f a, b;
        a.x = h[m0 * CB_ + k + khalf];
        a.y = h[m0 * CB_ + k + khalf + 1];
        // B[k][n] = w2[n][k]  (w2 is [512,32] row-major)
        b.x = w2[n0 * CB_ + k + khalf];
        b.y = w2[n0 * CB_ + k + khalf + 1];
        acc = __builtin_amdgcn_wmma_f32_16x16x4_f32(
            false, a, false, b, (short)0, acc, false, false);
    }

    const float bias = b2[n0];
    const int rbase  = mt * 16 + ((lane >> 4) << 3);
    #pragma unroll
    for (int r = 0; r < 8; ++r) {
        float v = acc[r] + bias;
        float gate = 1.0f / (1.0f + __expf(-v));
        const int m = rbase + r;
        out[m * C_ + n0] = g[m * C_ + n0] * gate;
    }
}

// ---------------------------------------------------------------------------
extern "C" void kernel_launch(void* const* d_in, const int* in_sizes, int n_in,
                              void* d_out, int out_size, void* d_ws, size_t ws_size,
                              hipStream_t stream) {
    const float* x        = (const float*)d_in[0];  // [64,512,56,56]
    const float* w1       = (const float*)d_in[1];  // [32,512]
    const float* b1       = (const float*)d_in[2];  // [32]
    const float* bn_gamma = (const float*)d_in[3];  // [32]
    const float* bn_beta  = (const float*)d_in[4];  // [32]
    const float* bn_mean  = (const float*)d_in[5];  // [32]
    const float* bn_var   = (const float*)d_in[6];  // [32]
    const float* w2       = (const float*)d_in[7];  // [512,32]
    const float* b2       = (const float*)d_in[8];  // [512]
    float* out = (float*)d_out;                     // [64,512]

    float* g = (float*)d_ws;                        // [64,512]  (128 KB)
    float* h = g + B_ * C_;                         // [64,32]

    // 1) streaming pooled reduction: 4096 blocks x 8 waves, 1 wave per row
    se_pool_kernel<<<(B_ * C_) / 8, 256, 0, stream>>>(x, g);
    // 2) squeeze MLP layer 1 + BN + ReLU (8 WMMA tiles, 1 block)
    se_mlp1_kernel<<<1, 256, 0, stream>>>(g, w1, b1, bn_gamma, bn_beta,
                                          bn_mean, bn_var, h);
    // 3) excite MLP layer 2 + sigmoid + gating (128 WMMA tiles)
    se_mlp2_kernel<<<16, 256, 0, stream>>>(h, w2, b2, g, out);
}